// MultiBoxLoss__45226005627591
// MI455X (gfx1250) — compile-verified
//
#include <hip/hip_runtime.h>
#include <hip/hip_bf16.h>
#include <stdint.h>

// ---------------- problem constants (match reference setup_inputs) ----------
#define B_      32
#define P_      131072
#define A_      32
#define TPB     256          // 8 waves (wave32)
#define NBLKB   64           // blocks per batch: 64*2048 = 131072 priors
#define PPT     8            // priors per thread
#define NBLK    (B_*NBLKB)   // 2048 total tiles

#define THRESH  0.35f
#define V0      0.1f
#define V1      0.2f

// ---------------- workspace layout (bytes) ----------------------------------
#define WS_PACKED   0                          // B_*A_ u64 (atomicMax argmax)
#define WS_POSMASK  (WS_PACKED + B_*A_*8)      // B_*(P_/32) u32
#define WS_PART_LL  (WS_POSMASK + B_*(P_/32)*4)
#define WS_PART_C5  (WS_PART_LL + NBLK*4)
#define WS_PART_CP  (WS_PART_C5 + NBLK*4)
#define WS_PART_NP  (WS_PART_CP + NBLK*4)
#define WS_STATE    (WS_PART_NP + NBLK*4)      // 132 ints, padded to 1KB
#define WS_HIST     (WS_STATE + 1024)          // 4 levels * B_ * 256 u32
#define WS_NEGP     (WS_HIST + 4*B_*256*4)     // NBLK floats
#define WS_LM       (WS_NEGP + NBLK*4)         // optional: B_*P_ f32 loss_mine cache (16 MB)
#define WS_SMALL    ((size_t)WS_LM)                       // ~0.69 MB, required
#define WS_BIG      ((size_t)WS_LM + (size_t)B_*P_*4)     // ~17.5 MB, enables L2-resident cache
// state indices (int units)
#define ST_K      0
#define ST_KREM   32
#define ST_CGT    64
#define ST_PREFIX 96
#define ST_TOT    128   // [0]=ll f32 [1]=c5 f32 [2]=cpos f32 [3]=np i32

// ---------------- CDNA5 async global->LDS staging ----------------------------
#if defined(__has_builtin)
# if __has_builtin(__builtin_amdgcn_global_load_async_to_lds_b32)
#  define HAVE_ASYNC_BUILTIN 1
# endif
# if __has_builtin(__builtin_amdgcn_s_wait_asynccnt)
#  define HAVE_WAIT_ASYNC_BUILTIN 1
# endif
#endif

__device__ __forceinline__ void wait_asynccnt0() {
#ifdef HAVE_WAIT_ASYNC_BUILTIN
  __builtin_amdgcn_s_wait_asynccnt(0);
#else
  asm volatile("s_wait_asynccnt 0" ::: "memory");
#endif
}

// Stage the 32x15-float per-batch target table into LDS via the CDNA5
// async-to-LDS path (GLOBAL_LOAD_ASYNC_TO_LDS_B32, ASYNCcnt).
__device__ __forceinline__ void stage_targets(const float* __restrict__ tg,
                                              float* struth, int tid) {
#ifdef HAVE_ASYNC_BUILTIN
  typedef __attribute__((address_space(1))) int as1_int;
  typedef __attribute__((address_space(3))) int as3_int;
  for (int e = tid; e < A_*15; e += TPB) {
    __builtin_amdgcn_global_load_async_to_lds_b32(
        (as1_int*)(uintptr_t)(tg + e),
        (as3_int*)(uintptr_t)(struth + e),   // flat LDS aperture: low 32 bits give the LDS offset
        0, 0);
  }
  wait_asynccnt0();
#else
  for (int e = tid; e < A_*15; e += TPB) {
    uint32_t ldsoff = (uint32_t)(uintptr_t)(struth + e);
    const float* g = tg + e;
    asm volatile("global_load_async_to_lds_b32 %0, %1, off"
                 :: "v"(ldsoff), "v"(g) : "memory");
  }
  wait_asynccnt0();
#endif
  __syncthreads();
}

// ---------------- helpers ----------------------------------------------------
__device__ __forceinline__ float sl1f(float d) {
  float a = fabsf(d);
  return (a < 1.f) ? 0.5f*d*d : a - 0.5f;
}
__device__ __forceinline__ float lse2(float a, float b) {
  float m = fmaxf(a, b);
  return m + logf(expf(a - m) + expf(b - m));
}
__device__ __forceinline__ float block_sum(float v, float* sred, int tid) {
  sred[tid] = v; __syncthreads();
  for (int s = TPB/2; s > 0; s >>= 1) { if (tid < s) sred[tid] += sred[tid+s]; __syncthreads(); }
  float r = sred[0]; __syncthreads(); return r;
}
__device__ __forceinline__ int block_sumi(int v, int* sred, int tid) {
  sred[tid] = v; __syncthreads();
  for (int s = TPB/2; s > 0; s >>= 1) { if (tid < s) sred[tid] += sred[tid+s]; __syncthreads(); }
  int r = sred[0]; __syncthreads(); return r;
}
__device__ __forceinline__ unsigned long long shfl_xor_u64(unsigned long long v, int m) {
  unsigned lo = (unsigned)v, hi = (unsigned)(v >> 32);
  lo = (unsigned)__shfl_xor((int)lo, m, 32);
  hi = (unsigned)__shfl_xor((int)hi, m, 32);
  return ((unsigned long long)hi << 32) | lo;
}

// ---------------- kernels ----------------------------------------------------
__global__ void k_zero(unsigned* p, int n) {
  int i = blockIdx.x*blockDim.x + threadIdx.x;
  if (i < n) p[i] = 0u;
}

// Pass A: per (batch, truth) argmax over priors of IoU, via packed u64 atomicMax.
__global__ void k_best_prior(const float* __restrict__ priors,
                             const float* __restrict__ targets,
                             unsigned long long* __restrict__ packed) {
  __shared__ float struth[A_*15];
  __shared__ float ta[A_];
  const int b = blockIdx.y, blk = blockIdx.x, tid = threadIdx.x;
  stage_targets(targets + (size_t)b*A_*15, struth, tid);
  if (tid < A_) {
    const float* t = struth + tid*15;
    ta[tid] = (t[2]-t[0])*(t[3]-t[1]);
  }
  __syncthreads();

  const int base = blk*(TPB*PPT);
  float px1[PPT], py1[PPT], px2[PPT], py2[PPT], pa[PPT];
  for (int j = 0; j < PPT; j++) {
    int p = base + j*TPB + tid;
    float4 pr = *(const float4*)(priors + (size_t)p*4);   // cx,cy,w,h
    px1[j] = pr.x - pr.z*0.5f; py1[j] = pr.y - pr.w*0.5f;
    px2[j] = pr.x + pr.z*0.5f; py2[j] = pr.y + pr.w*0.5f;
    pa[j]  = (px2[j]-px1[j])*(py2[j]-py1[j]);
  }
  const int lane = tid & 31;
  for (int a = 0; a < A_; a++) {
    const float* t = struth + a*15;
    float tx1=t[0], ty1=t[1], tx2=t[2], ty2=t[3], aa=ta[a];
    unsigned long long best = 0ull;
    for (int j = 0; j < PPT; j++) {
      int p = base + j*TPB + tid;
      float w = fmaxf(fminf(tx2, px2[j]) - fmaxf(tx1, px1[j]), 0.f);
      float h = fmaxf(fminf(ty2, py2[j]) - fmaxf(ty1, py1[j]), 0.f);
      float inter = w*h;
      float iou = inter / (aa + pa[j] - inter);
      unsigned long long pk = ((unsigned long long)__float_as_uint(iou) << 32)
                              | (unsigned)(~(unsigned)p);   // ties -> smallest index
      best = (pk > best) ? pk : best;
    }
    for (int off = 16; off > 0; off >>= 1) {                // wave32 reduce
      unsigned long long o = shfl_xor_u64(best, off);
      best = (o > best) ? o : best;
    }
    if (lane == 0) atomicMax(&packed[b*A_ + a], best);
  }
}

// Pass B: match each prior, emit wave32-ballot pos bitmask, accumulate positive-part
// sums, and (optionally) cache loss_mine for the mining passes (L2-resident, 16 MB).
__global__ void k_match(const float* __restrict__ priors,
                        const float* __restrict__ targets,
                        const float* __restrict__ loc,
                        const float* __restrict__ conf,
                        const float* __restrict__ loc5,
                        const unsigned long long* __restrict__ packed,
                        unsigned* __restrict__ posmask,
                        float* __restrict__ part_ll, float* __restrict__ part_c5,
                        float* __restrict__ part_cp, int* __restrict__ part_np,
                        float* __restrict__ lm_out /* may be null */) {
  __shared__ float struth[A_*15];
  __shared__ float ta[A_];
  __shared__ int   bpi[A_];
  __shared__ float sred[TPB];
  __shared__ int   sredi[TPB];
  const int b = blockIdx.y, blk = blockIdx.x, tid = threadIdx.x;
  stage_targets(targets + (size_t)b*A_*15, struth, tid);
  if (tid < A_) {
    const float* t = struth + tid*15;
    ta[tid]  = (t[2]-t[0])*(t[3]-t[1]);
    bpi[tid] = (int)(~(unsigned)(packed[b*A_ + tid] & 0xFFFFFFFFull));
  }
  __syncthreads();

  const bool cache_lm = (lm_out != nullptr);
  float ll = 0.f, c5 = 0.f, cp = 0.f; int np = 0;
  const int base = blk*(TPB*PPT);
  for (int j = 0; j < PPT; j++) {
    const int p = base + j*TPB + tid;
    float4 pr = *(const float4*)(priors + (size_t)p*4);
    float px1 = pr.x - pr.z*0.5f, py1 = pr.y - pr.w*0.5f;
    float px2 = pr.x + pr.z*0.5f, py2 = pr.y + pr.w*0.5f;
    float pa  = (px2-px1)*(py2-py1);

    float bto = -1.f; int bti = 0;
    for (int a = 0; a < A_; a++) {
      const float* t = struth + a*15;
      float w = fmaxf(fminf(t[2], px2) - fmaxf(t[0], px1), 0.f);
      float h = fmaxf(fminf(t[3], py2) - fmaxf(t[1], py1), 0.f);
      float inter = w*h;
      float iou = inter / (ta[a] + pa - inter);
      if (iou > bto) { bto = iou; bti = a; }                // first-max tiebreak
    }
    for (int a = 0; a < A_; a++) if (bpi[a] == p) { bto = 2.f; bti = a; }

    const float* t = struth + bti*15;
    int cls = (bto < THRESH) ? 0 : (int)t[4];
    bool pos = cls > 0;
    unsigned m32 = (unsigned)__ballot(pos);                 // wave32: exact mask word
    if ((tid & 31) == 0) posmask[(size_t)b*(P_/32) + (p >> 5)] = m32;

    const size_t row = (size_t)b*P_ + p;
    if (cache_lm) {
      float2 cd = *(const float2*)(conf + row*2);
      float lse = lse2(cd.x, cd.y);
      lm_out[row] = pos ? 0.f : (lse - cd.x);               // loss_mine
      if (pos) {
        float gt = (cls == 1) ? cd.y : cd.x;
        cp += lse - gt; np++;
      }
    } else if (pos) {
      float2 cd = *(const float2*)(conf + row*2);
      float gt = (cls == 1) ? cd.y : cd.x;
      cp += lse2(cd.x, cd.y) - gt; np++;
    }

    if (pos) {
      float4 ld = *(const float4*)(loc + row*4);
      float gcx = ((t[0]+t[2])*0.5f - pr.x) / (V0*pr.z);
      float gcy = ((t[1]+t[3])*0.5f - pr.y) / (V0*pr.w);
      float gw  = logf((t[2]-t[0]) / pr.z) / V1;
      float gh  = logf((t[3]-t[1]) / pr.w) / V1;
      ll += sl1f(ld.x-gcx) + sl1f(ld.y-gcy) + sl1f(ld.z-gw) + sl1f(ld.w-gh);
      const float* l5 = loc5 + row*10;
      for (int q = 0; q < 5; q++) {
        float gx = (t[5+2*q] - pr.x) / (V0*pr.z);
        float gy = (t[6+2*q] - pr.y) / (V0*pr.w);
        float dx = l5[2*q] - gx, dy = l5[2*q+1] - gy;
        c5 += dx*dx + dy*dy;
      }
    }
  }
  float rll = block_sum(ll, sred, tid);
  float rc5 = block_sum(c5, sred, tid);
  float rcp = block_sum(cp, sred, tid);
  int   rnp = block_sumi(np, sredi, tid);
  if (tid == 0) {
    int g = b*NBLKB + blk;
    part_ll[g] = rll; part_c5[g] = rc5; part_cp[g] = rcp; part_np[g] = rnp;
  }
}

// Deterministic reduce of tile partials; init per-batch radix-select state.
__global__ void k_reduce(const float* __restrict__ part_ll, const float* __restrict__ part_c5,
                         const float* __restrict__ part_cp, const int* __restrict__ part_np,
                         int* __restrict__ st) {
  __shared__ float sred[TPB];
  __shared__ int   sredi[TPB];
  const int tid = threadIdx.x;
  if (tid < B_) {
    int np = 0;
    for (int i = 0; i < NBLKB; i++) np += part_np[tid*NBLKB + i];
    int k = 7*np; if (k > P_-1) k = P_-1;
    st[ST_K+tid] = k; st[ST_KREM+tid] = k; st[ST_CGT+tid] = 0;
    ((unsigned*)st)[ST_PREFIX+tid] = 0u;
  }
  float a = 0.f, c = 0.f, d = 0.f; int n = 0;
  for (int i = tid; i < NBLK; i += TPB) {
    a += part_ll[i]; c += part_c5[i]; d += part_cp[i]; n += part_np[i];
  }
  float ra = block_sum(a, sred, tid);
  float rc = block_sum(c, sred, tid);
  float rd = block_sum(d, sred, tid);
  int   rn = block_sumi(n, sredi, tid);
  if (tid == 0) {
    ((float*)st)[ST_TOT+0] = ra; ((float*)st)[ST_TOT+1] = rc;
    ((float*)st)[ST_TOT+2] = rd; st[ST_TOT+3] = rn;
  }
}

// loss_mine for one prior, recompute variant.
__device__ __forceinline__ float lm_recompute(const float* conf, const unsigned* posmask,
                                              int b, int p, int tid) {
  const unsigned word = posmask[(size_t)b*(P_/32) + (p >> 5)];
  const bool pos = (word >> (tid & 31)) & 1u;
  if (pos) return 0.f;
  float2 cd = *(const float2*)(conf + ((size_t)b*P_ + p)*2);
  return lse2(cd.x, cd.y) - cd.x;
}

// Radix-select histogram over loss_mine bits (level = byte 0..3, MSB first).
template <bool CACHED>
__global__ void k_hist(const float* __restrict__ conf, const unsigned* __restrict__ posmask,
                       const float* __restrict__ lm_in,
                       const int* __restrict__ st, unsigned* __restrict__ hist, int level) {
  __shared__ unsigned sh[256];
  const int b = blockIdx.y, blk = blockIdx.x, tid = threadIdx.x;
  sh[tid] = 0u; __syncthreads();
  const int k = st[ST_K+b];
  const unsigned pref = ((const unsigned*)st)[ST_PREFIX+b];
  if (k > 0) {
    const int base = blk*(TPB*PPT);
    for (int j = 0; j < PPT; j++) {
      const int p = base + j*TPB + tid;
      if (j + 1 < PPT) {
        if (CACHED) __builtin_prefetch(lm_in + ((size_t)b*P_ + (p + TPB)), 0, 1);
        else        __builtin_prefetch(conf + ((size_t)b*P_ + (p + TPB))*2, 0, 1);
      }
      float lm = CACHED ? lm_in[(size_t)b*P_ + p]
                        : lm_recompute(conf, posmask, b, p, tid);
      const unsigned bits = __float_as_uint(lm);            // lm >= 0: order-preserving bits
      bool ok = true;
      if (level > 0) { int s = 32 - 8*level; ok = ((bits >> s) == (pref >> s)); }
      if (ok) atomicAdd(&sh[(bits >> (24 - 8*level)) & 0xFFu], 1u);
    }
  }
  __syncthreads();
  if (sh[tid]) atomicAdd(&hist[((size_t)level*B_ + b)*256 + tid], sh[tid]);
}

// Pick the byte bucket containing the k-th largest; update prefix/krem/count_gt.
__global__ void k_select(const unsigned* __restrict__ hist, int* __restrict__ st, int level) {
  const int b = threadIdx.x;
  if (b >= B_) return;
  if (st[ST_K+b] <= 0) return;
  const unsigned* h = hist + ((size_t)level*B_ + b)*256;
  const int krem = st[ST_KREM+b];
  long long cum = 0; int sel = 0;
  for (int bin = 255; bin >= 0; bin--) {
    unsigned c = h[bin];
    if (cum + (long long)c >= (long long)krem) { sel = bin; break; }
    cum += c;
  }
  st[ST_CGT+b]  += (int)cum;
  st[ST_KREM+b]  = krem - (int)cum;
  ((unsigned*)st)[ST_PREFIX+b] |= (unsigned)sel << (24 - 8*level);
}

// Sum loss_mine strictly above the per-batch k-th-largest threshold.
template <bool CACHED>
__global__ void k_negsum(const float* __restrict__ conf, const unsigned* __restrict__ posmask,
                         const float* __restrict__ lm_in,
                         const int* __restrict__ st, float* __restrict__ negp) {
  __shared__ float sred[TPB];
  const int b = blockIdx.y, blk = blockIdx.x, tid = threadIdx.x;
  const int k = st[ST_K+b];
  const unsigned tb = (k > 0) ? ((const unsigned*)st)[ST_PREFIX+b] : 0xFFFFFFFFu;
  float s = 0.f;
  const int base = blk*(TPB*PPT);
  for (int j = 0; j < PPT; j++) {
    const int p = base + j*TPB + tid;
    if (j + 1 < PPT) {
      if (CACHED) __builtin_prefetch(lm_in + ((size_t)b*P_ + (p + TPB)), 0, 1);
      else        __builtin_prefetch(conf + ((size_t)b*P_ + (p + TPB))*2, 0, 1);
    }
    float lm = CACHED ? lm_in[(size_t)b*P_ + p]
                      : lm_recompute(conf, posmask, b, p, tid);
    if (__float_as_uint(lm) > tb) s += lm;
  }
  float r = block_sum(s, sred, tid);
  if (tid == 0) negp[b*NBLKB + blk] = r;
}

__global__ void k_final(const float* __restrict__ negp, const int* __restrict__ st,
                        float* __restrict__ out) {
  __shared__ float sred[TPB];
  const int tid = threadIdx.x;
  float s = 0.f;
  for (int i = tid; i < NBLK; i += TPB) s += negp[i];
  float negsum = block_sum(s, sred, tid);
  if (tid == 0) {
    float tie = 0.f;
    for (int b = 0; b < B_; b++) {
      if (st[ST_K+b] > 0) {
        float t = __uint_as_float(((const unsigned*)st)[ST_PREFIX+b]); // exact k-th largest
        tie += (float)st[ST_KREM+b] * t;                               // ties at threshold
      }
    }
    float ll   = ((const float*)st)[ST_TOT+0];
    float c5   = ((const float*)st)[ST_TOT+1];
    float cpos = ((const float*)st)[ST_TOT+2];
    int   npt  = st[ST_TOT+3];
    float N = fmaxf((float)npt, 1.f);
    out[0] = ll / N;
    out[1] = (cpos + negsum + tie) / N;
    out[2] = c5 / N;
  }
}

// ---------------- launch ------------------------------------------------------
extern "C" void kernel_launch(void* const* d_in, const int* in_sizes, int n_in,
                              void* d_out, int out_size, void* d_ws, size_t ws_size,
                              hipStream_t stream) {
  const float* loc     = (const float*)d_in[0];  // (B,P,4)
  const float* conf    = (const float*)d_in[1];  // (B,P,2)
  const float* loc5    = (const float*)d_in[2];  // (B,P,10)
  const float* priors  = (const float*)d_in[3];  // (P,4)
  const float* targets = (const float*)d_in[4];  // (B,A,15)
  (void)in_sizes; (void)n_in; (void)out_size;

  char* ws = (char*)d_ws;   // requires ws_size >= WS_SMALL (~0.69 MB)
  unsigned long long* packed = (unsigned long long*)(ws + WS_PACKED);
  unsigned* posmask = (unsigned*)(ws + WS_POSMASK);
  float* part_ll = (float*)(ws + WS_PART_LL);
  float* part_c5 = (float*)(ws + WS_PART_C5);
  float* part_cp = (float*)(ws + WS_PART_CP);
  int*   part_np = (int*)(ws + WS_PART_NP);
  int*   st      = (int*)(ws + WS_STATE);
  unsigned* hist = (unsigned*)(ws + WS_HIST);
  float* negp    = (float*)(ws + WS_NEGP);
  const bool cached = (ws_size >= WS_BIG);       // fixed per run -> deterministic
  float* lm = cached ? (float*)(ws + WS_LM) : nullptr;

  k_zero<<<(B_*A_*2 + 255)/256, 256, 0, stream>>>((unsigned*)packed, B_*A_*2);
  k_zero<<<(4*B_*256 + 255)/256, 256, 0, stream>>>(hist, 4*B_*256);

  dim3 grid(NBLKB, B_);
  k_best_prior<<<grid, TPB, 0, stream>>>(priors, targets, packed);
  k_match<<<grid, TPB, 0, stream>>>(priors, targets, loc, conf, loc5,
                                    packed, posmask, part_ll, part_c5, part_cp, part_np, lm);
  k_reduce<<<1, TPB, 0, stream>>>(part_ll, part_c5, part_cp, part_np, st);
  for (int L = 0; L < 4; ++L) {
    if (cached) k_hist<true ><<<grid, TPB, 0, stream>>>(conf, posmask, lm, st, hist, L);
    else        k_hist<false><<<grid, TPB, 0, stream>>>(conf, posmask, lm, st, hist, L);
    k_select<<<1, 32, 0, stream>>>(hist, st, L);
  }
  if (cached) k_negsum<true ><<<grid, TPB, 0, stream>>>(conf, posmask, lm, st, negp);
  else        k_negsum<false><<<grid, TPB, 0, stream>>>(conf, posmask, lm, st, negp);
  k_final<<<1, TPB, 0, stream>>>(negp, st, (float*)d_out);
}